// Zpool2D_Window_42056319762980
// MI455X (gfx1250) — compile-verified
//
#include <hip/hip_runtime.h>
#include <math.h>

#define HGT 384
#define WID 384
#define NPIX (HGT * WID)          // 147456
#define RING 60
#define NWAVES 12
#define NPLANES 128               // B*C = 2*64

typedef __attribute__((ext_vector_type(2))) float v2f;
typedef __attribute__((ext_vector_type(4))) float v4f;
typedef __attribute__((ext_vector_type(8))) float v8f;

__device__ __forceinline__ int imin(int a, int b) { return a < b ? a : b; }
__device__ __forceinline__ int imax(int a, int b) { return a > b ? a : b; }

// One wave computes horizontal box sums (windows 7/15/31) for row r of the plane,
// writing three 384-float rows into the LDS ring at slot r%RING.
__device__ __forceinline__ void hsum_row(const float* __restrict__ xplane,
                                         float* __restrict__ Hs,
                                         float* __restrict__ Prow,   // 384 floats, per-wave
                                         int r, int lane)
{
    const float* xrow = xplane + r * WID;
    const int c0 = lane * 12;

    // in-lane inclusive prefix over 12 columns
    float p[12];
    float run = 0.f;
#pragma unroll
    for (int i = 0; i < 12; ++i) { run += xrow[c0 + i]; p[i] = run; }

    // wave-level exclusive scan of lane totals (wave32)
    float tot = run;
    float sc = tot;
#pragma unroll
    for (int d = 1; d < 32; d <<= 1) {
        float o = __shfl_up(sc, (unsigned)d, 32);
        if (lane >= d) sc += o;
    }
    const float excl = sc - tot;

#pragma unroll
    for (int i = 0; i < 12; ++i) Prow[c0 + i] = p[i] + excl;
    // same-wave cross-lane LDS RAW: wait for our DS writes to land
    __asm__ volatile("s_wait_dscnt 0" ::: "memory");

    const int slot = r % RING;
    float* h7  = Hs + (0 * RING + slot) * WID;
    float* h15 = Hs + (1 * RING + slot) * WID;
    float* h31 = Hs + (2 * RING + slot) * WID;

#pragma unroll
    for (int i = 0; i < 12; ++i) {
        const int c = c0 + i;
        const float Ph3  = Prow[imin(c + 3,  WID - 1)];
        const float Ph7  = Prow[imin(c + 7,  WID - 1)];
        const float Ph15 = Prow[imin(c + 15, WID - 1)];
        const float Pl4  = (c >= 4)  ? Prow[c - 4]  : 0.f;
        const float Pl8  = (c >= 8)  ? Prow[c - 8]  : 0.f;
        const float Pl16 = (c >= 16) ? Prow[c - 16] : 0.f;
        h7[c]  = Ph3  - Pl4;
        h15[c] = Ph7  - Pl8;
        h31[c] = Ph15 - Pl16;
    }
}

// One workgroup per (b,c) plane: sliding separable box filter in LDS ring,
// accumulates Σf, Σf² for 3 windows + Σx, Σx², reduces via F32 WMMA,
// writes 4 inverse-stds per plane.
__global__ __launch_bounds__(384, 1)
void zpool_stats_kernel(const float* __restrict__ x,
                        const float* __restrict__ tinyw,
                        float* __restrict__ invstd)
{
    extern __shared__ float smem[];
    float* Hs   = smem;                      // 3*RING*WID   = 69120 f
    float* Prow = smem + 3 * RING * WID;     // NWAVES*WID   =  4608 f
    float* red  = Prow + NWAVES * WID;       // NWAVES*8     =    96 f

    const int tid   = threadIdx.x;
    const int lane  = tid & 31;
    const int wave  = tid >> 5;
    const int plane = blockIdx.x;
    const float* xplane = x + (size_t)plane * NPIX;

    // prologue: horizontal sums for rows 0..26 (covers first block's vertical needs)
    for (int r = wave; r < 27; r += NWAVES)
        hsum_row(xplane, Hs, Prow + wave * WID, r, lane);
    __syncthreads();

    const int c = tid;                       // this thread owns column c
    const int w2arr[3] = {3, 7, 15};

    float inv_cw[3];
#pragma unroll
    for (int w = 0; w < 3; ++w) {
        const int w2 = w2arr[w];
        inv_cw[w] = 1.0f / (float)(imin(c + w2, WID - 1) - imax(c - w2, 0) + 1);
    }

    // vertical running sums primed for output row 0 (rows 0..w2-1 pre-added)
    float V[3];
#pragma unroll
    for (int w = 0; w < 3; ++w) {
        const int w2 = w2arr[w];
        float v = 0.f;
        for (int k = 0; k < w2; ++k) v += Hs[(w * RING + k) * WID + c];
        V[w] = v;
    }

    float s[4] = {0.f, 0.f, 0.f, 0.f};
    float q[4] = {0.f, 0.f, 0.f, 0.f};

    for (int j = 0; j < 32; ++j) {
        // produce next block's hsum rows (12j+27 .. 12j+38); slots are dead by ring analysis
        const int hr = 12 * j + 27 + wave;
        if (hr < HGT) hsum_row(xplane, Hs, Prow + wave * WID, hr, lane);
        __syncthreads();

        for (int stepi = 0; stepi < 12; ++stepi) {
            const int t = 12 * j + stepi;
            const float xv = xplane[t * WID + c];
            s[3] += xv;
            q[3] += xv * xv;
#pragma unroll
            for (int w = 0; w < 3; ++w) {
                const int w2 = w2arr[w];
                const int re = t + w2;
                if (re < HGT) V[w] += Hs[(w * RING + (re % RING)) * WID + c];
                const int ch = imin(t + w2, HGT - 1) - imax(t - w2, 0) + 1;
                const float mean = V[w] * inv_cw[w] / (float)ch;
                const float f = xv - mean;
                s[w] += f;
                q[w] += f * f;
                const int rl = t - w2;
                if (rl >= 0) V[w] -= Hs[(w * RING + (rl % RING)) * WID + c];
            }
        }
        __syncthreads();   // consumption done before next block overwrites ring slots
    }

    // ---- reduction: 8 quantities across 384 threads ----
    float acc[8] = {s[0], s[1], s[2], s[3], q[0], q[1], q[2], q[3]};
#pragma unroll
    for (int qi = 0; qi < 8; ++qi) {
        float v = acc[qi];
#pragma unroll
        for (int m = 16; m >= 1; m >>= 1) v += __shfl_xor(v, m, 32);
        acc[qi] = v;
    }
    if (lane == 0) {
#pragma unroll
        for (int qi = 0; qi < 8; ++qi) red[wave * 8 + qi] = acc[qi];
    }
    __syncthreads();

    if (wave == 0) {
        // Pack 12 wave-partials as A[16x4] (M=quantity, K=4 groups of 3 waves),
        // B = ones -> D[q][n] = total of quantity q. F32 WMMA keeps full precision.
        const int qi = lane & 15;
        const bool hi = lane >= 16;          // lanes 16-31 hold K=2 / K=3 per ISA layout
        float a0 = 0.f, a1 = 0.f;
        if (qi < 8) {
            const int wb0 = hi ? 6 : 0;      // K = 2 : 0  -> waves 3K..3K+2
            const int wb1 = hi ? 9 : 3;      // K = 3 : 1
            a0 = red[(wb0 + 0) * 8 + qi] + red[(wb0 + 1) * 8 + qi] + red[(wb0 + 2) * 8 + qi];
            a1 = red[(wb1 + 0) * 8 + qi] + red[(wb1 + 1) * 8 + qi] + red[(wb1 + 2) * 8 + qi];
        }
        v2f A; A[0] = a0;  A[1] = a1;
        v2f B; B[0] = 1.f; B[1] = 1.f;
        v8f C = {};
        v8f D = __builtin_amdgcn_wmma_f32_16x16x4_f32(
            false, A, false, B, (short)0, C, false, false);

        if (lane == 0) {
            const float Nn = (float)NPIX;
            const float rN = 1.0f / Nn;
            const float rD = 1.0f / (Nn - 1.0f);
            const float tiny = tinyw[plane & 63] + 1e-6f;   // tiny_weight[c] + MIN/10
#pragma unroll
            for (int l = 0; l < 4; ++l) {
                const float S = D[l];
                const float Q = D[4 + l];
                float var = (Q - S * S * rN) * rD;           // ddof=1
                var = fmaxf(var, 0.f);
                const float sd = sqrtf(var);
                invstd[plane * 4 + l] = 1.0f / fmaxf(sd, tiny);
            }
        }
    }
}

// Streaming pass: out[b,l,c,:,:] = x[b,c,:,:] * invstd[b,l,c]. 302 MB of NT stores.
__global__ __launch_bounds__(256)
void zpool_scale_kernel(const float* __restrict__ x,
                        const float* __restrict__ invstd,
                        float* __restrict__ out)
{
    const int plane = blockIdx.x / 144;      // 144 blocks per plane (1024 px each)
    const int chunk = blockIdx.x % 144;
    const int b = plane >> 6;
    const int cc = plane & 63;

    const float i0 = invstd[plane * 4 + 0];
    const float i1 = invstd[plane * 4 + 1];
    const float i2 = invstd[plane * 4 + 2];
    const float i3 = invstd[plane * 4 + 3];

    const int pix = chunk * 1024 + threadIdx.x * 4;
    const v4f xv = *reinterpret_cast<const v4f*>(x + (size_t)plane * NPIX + pix);

    const size_t lstride = (size_t)64 * NPIX;
    const size_t base0 = ((size_t)b * 256 + (size_t)cc) * NPIX + (size_t)pix;

    __builtin_nontemporal_store(xv * i0, reinterpret_cast<v4f*>(out + base0));
    __builtin_nontemporal_store(xv * i1, reinterpret_cast<v4f*>(out + base0 + lstride));
    __builtin_nontemporal_store(xv * i2, reinterpret_cast<v4f*>(out + base0 + 2 * lstride));
    __builtin_nontemporal_store(xv * i3, reinterpret_cast<v4f*>(out + base0 + 3 * lstride));
}

extern "C" void kernel_launch(void* const* d_in, const int* in_sizes, int n_in,
                              void* d_out, int out_size, void* d_ws, size_t ws_size,
                              hipStream_t stream)
{
    const float* x     = (const float*)d_in[0];   // (2,64,384,384) f32
    const float* tinyw = (const float*)d_in[1];   // (1,1,64,1,1) f32
    float* out    = (float*)d_out;                // (2,4,64,384,384) f32
    float* invstd = (float*)d_ws;                 // 128*4 floats scratch

    const size_t smem_bytes =
        (size_t)(3 * RING * WID + NWAVES * WID + NWAVES * 8) * sizeof(float); // 295296 B

    zpool_stats_kernel<<<NPLANES, 384, smem_bytes, stream>>>(x, tinyw, invstd);
    zpool_scale_kernel<<<NPLANES * 144, 256, 0, stream>>>(x, invstd, out);
}